// RegistrationCriterion_65962107732766
// MI455X (gfx1250) — compile-verified
//
#include <hip/hip_runtime.h>
#include <hip/hip_bf16.h>

typedef float v2f __attribute__((ext_vector_type(2)));
typedef float v8f __attribute__((ext_vector_type(8)));

#define N_VOX  (128*128*128)   /* 2^21 = 2097152 */
#define NMASK  (N_VOX - 1)

// -----------------------------------------------------------------------------
// ws layout (floats):
//   [0..63]   accumulators: 0=lncc_sum 1=grad_sum 2=jac_sum 3=inv_sum
//   [64 ..)   xf: 5 * N_VOX   (x-filtered fields s,t,ss,tt,st)
//   then      yf: 5 * N_VOX   (xy-filtered fields)
// total required: (64 + 10*N_VOX)*4 bytes ~= 84 MB (L2-resident on MI455X)
// -----------------------------------------------------------------------------

__global__ __launch_bounds__(64) void k_init(float* acc) {
    if (threadIdx.x < 64) acc[threadIdx.x] = 0.0f;
}

// -----------------------------------------------------------------------------
// Stage 1: x-direction 5-tap box sum of the five LNCC statistic fields,
// expressed as a banded matmul on the WMMA pipe:
//   Out[r, j] = sum_{d=0..19} In[r, x0-2+d] * Band[d, j],  Band[d,j]=1 iff j<=d<=j+4
// 5 chunks of K=4 per 16x16 tile, v_wmma_f32_16x16x4_f32, f32-exact.
// Rows r = z*128+y (16384 rows), 8 x-tiles per row. One tile per wave32.
// -----------------------------------------------------------------------------
__global__ __launch_bounds__(256) void k_lncc_xfilter_wmma(const float* __restrict__ s,
                                                           const float* __restrict__ t,
                                                           float* __restrict__ xf) {
    const int wave = threadIdx.x >> 5;
    const int lane = threadIdx.x & 31;
    const int task = blockIdx.x * 8 + wave;      // 1024 blocks * 8 waves = 8192 tiles
    const int rowTile = task >> 3;               // 0..1023
    const int xTile   = task & 7;                // 0..7
    const int rowBase = rowTile * 16;
    const int x0      = xTile * 16;

    const int m  = lane & 15;                    // A-matrix row within tile
    const int kb = (lane >> 4) * 2;              // lanes 0-15 -> K {0,1}, 16-31 -> K {2,3}
    const int row = rowBase + m;
    const int rb  = row * 128;

    v8f cS = {}; v8f cT = {}; v8f cSS = {}; v8f cTT = {}; v8f cST = {};

#pragma unroll
    for (int c = 0; c < 5; ++c) {
        const int d0  = 4 * c + kb;              // band row index of this lane's K=0 slot
        const int d1  = d0 + 1;
        const int ix0 = x0 - 2 + d0;             // input x positions (zero-padded)
        const int ix1 = ix0 + 1;
        const bool v0 = (ix0 >= 0) & (ix0 < 128);
        const bool v1 = (ix1 >= 0) & (ix1 < 128);
        const float s0 = v0 ? s[rb + ix0] : 0.0f;
        const float t0 = v0 ? t[rb + ix0] : 0.0f;
        const float s1 = v1 ? s[rb + ix1] : 0.0f;
        const float t1 = v1 ? t[rb + ix1] : 0.0f;

        v2f aS  = { s0,      s1      };
        v2f aT  = { t0,      t1      };
        v2f aSS = { s0 * s0, s1 * s1 };
        v2f aTT = { t0 * t0, t1 * t1 };
        v2f aST = { s0 * t0, s1 * t1 };

        // B fragment: lane's column n = lane&15; B[d,n] = 1 iff n <= d <= n+4
        const int n = m;
        v2f b;
        b.x = (n <= d0 && d0 <= n + 4) ? 1.0f : 0.0f;
        b.y = (n <= d1 && d1 <= n + 4) ? 1.0f : 0.0f;

        cS  = __builtin_amdgcn_wmma_f32_16x16x4_f32(false, aS,  false, b, (short)0, cS,  false, false);
        cT  = __builtin_amdgcn_wmma_f32_16x16x4_f32(false, aT,  false, b, (short)0, cT,  false, false);
        cSS = __builtin_amdgcn_wmma_f32_16x16x4_f32(false, aSS, false, b, (short)0, cSS, false, false);
        cTT = __builtin_amdgcn_wmma_f32_16x16x4_f32(false, aTT, false, b, (short)0, cTT, false, false);
        cST = __builtin_amdgcn_wmma_f32_16x16x4_f32(false, aST, false, b, (short)0, cST, false, false);
    }

    // D layout: VGPR g -> M = g (lanes 0-15, N=lane) and M = g+8 (lanes 16-31, N=lane-16)
    const int half = lane >> 4;
    const int nCol = lane & 15;
#pragma unroll
    for (int g = 0; g < 8; ++g) {
        const int mRow = g + half * 8;
        const int o = (rowBase + mRow) * 128 + x0 + nCol;
        xf[0 * N_VOX + o] = cS[g];
        xf[1 * N_VOX + o] = cT[g];
        xf[2 * N_VOX + o] = cSS[g];
        xf[3 * N_VOX + o] = cTT[g];
        xf[4 * N_VOX + o] = cST[g];
    }
}

// -----------------------------------------------------------------------------
// Stage 2: y-direction 5-tap box sum (zero pad). Lanes are consecutive in x
// so every load/store is fully coalesced; traffic stays in L2.
// -----------------------------------------------------------------------------
__global__ __launch_bounds__(256) void k_lncc_yfilter(const float* __restrict__ xf,
                                                      float* __restrict__ yf) {
    const int tid = blockIdx.x * 256 + threadIdx.x;      // 5*N_VOX threads
    const int f = tid >> 21;                             // field
    const int v = tid & NMASK;
    const int x = v & 127;
    const int y = (v >> 7) & 127;
    const int z = v >> 14;
    const float* in = xf + f * N_VOX + (z << 14) + x;
    float a = 0.0f;
#pragma unroll
    for (int k = -2; k <= 2; ++k) {
        const int yy = y + k;
        if (yy >= 0 && yy < 128) a += in[yy << 7];
    }
    yf[tid] = a;
}

// -----------------------------------------------------------------------------
// Stage 3: z-direction box sum + per-voxel LNCC + block reduction.
// -----------------------------------------------------------------------------
__global__ __launch_bounds__(256) void k_lncc_final(const float* __restrict__ yf,
                                                    float* __restrict__ acc) {
    const int v = blockIdx.x * 256 + threadIdx.x;
    const int x = v & 127;
    const int y = (v >> 7) & 127;
    const int z = v >> 14;

    float S[5];
#pragma unroll
    for (int f = 0; f < 5; ++f) {
        const float* in = yf + f * N_VOX + (y << 7) + x;
        float a = 0.0f;
#pragma unroll
        for (int k = -2; k <= 2; ++k) {
            const int zz = z + k;
            if (zz >= 0 && zz < 128) a += in[zz << 14];
        }
        S[f] = a * (1.0f / 125.0f);
    }
    const float sm = S[0], tm = S[1];
    const float sv = S[2] - sm * sm;
    const float tv = S[3] - tm * tm;
    const float cr = S[4] - sm * tm;
    const float lncc = cr * cr / (sv * tv + 1e-5f);

    __shared__ float red[256];
    red[threadIdx.x] = lncc;
    __syncthreads();
    for (int off = 128; off > 0; off >>= 1) {
        if (threadIdx.x < off) red[threadIdx.x] += red[threadIdx.x + off];
        __syncthreads();
    }
    if (threadIdx.x == 0) atomicAdd(&acc[0], red[0]);
}

// -----------------------------------------------------------------------------
// Fused smoothness (grad L2) + negative-Jacobian over forward_disp.
// One pass: 12 loads/voxel shared between both terms.
// -----------------------------------------------------------------------------
__global__ __launch_bounds__(256) void k_gradjac(const float* __restrict__ d,
                                                 float* __restrict__ acc) {
    const int v = blockIdx.x * 256 + threadIdx.x;
    const int x = v & 127;
    const int y = (v >> 7) & 127;
    const int z = v >> 14;
    const bool hz = z < 127, hy = y < 127, hx = x < 127;

    float c0[3], zp[3], yp[3], xp[3];
#pragma unroll
    for (int ch = 0; ch < 3; ++ch) {
        const int b = ch * N_VOX + v;
        c0[ch] = d[b];
        zp[ch] = hz ? d[b + (1 << 14)] : 0.0f;
        yp[ch] = hy ? d[b + 128]       : 0.0f;
        xp[ch] = hx ? d[b + 1]         : 0.0f;
    }

    float g = 0.0f;
#pragma unroll
    for (int ch = 0; ch < 3; ++ch) {
        if (hz) { const float dd = zp[ch] - c0[ch]; g += dd * dd; }
        if (hy) { const float dd = yp[ch] - c0[ch]; g += dd * dd; }
        if (hx) { const float dd = xp[ch] - c0[ch]; g += dd * dd; }
    }

    float j = 0.0f;
    if (hz & hy & hx) {
        const float dx0 = zp[0] - c0[0], dx1 = zp[1] - c0[1], dx2 = zp[2] - c0[2];
        const float dy0 = yp[0] - c0[0], dy1 = yp[1] - c0[1], dy2 = yp[2] - c0[2];
        const float dz0 = xp[0] - c0[0], dz1 = xp[1] - c0[1], dz2 = xp[2] - c0[2];
        const float a00 = 1.0f + dx0, a01 = dy0, a02 = dz0;
        const float a10 = dx1, a11 = 1.0f + dy1, a12 = dz1;
        const float a20 = dx2, a21 = dy2, a22 = 1.0f + dz2;
        float det = a00 * (a11 * a22 - a12 * a21)
                  - a01 * (a10 * a22 - a12 * a20)
                  + a02 * (a10 * a21 - a11 * a20);
        if (isnan(det)) det = 0.0f;
        else if (isinf(det)) det = det > 0.0f ? 1000.0f : -1000.0f;
        j = fmaxf(-det, 0.0f);
    }

    __shared__ float rg[256];
    __shared__ float rj[256];
    rg[threadIdx.x] = g;
    rj[threadIdx.x] = j;
    __syncthreads();
    for (int off = 128; off > 0; off >>= 1) {
        if (threadIdx.x < off) {
            rg[threadIdx.x] += rg[threadIdx.x + off];
            rj[threadIdx.x] += rj[threadIdx.x + off];
        }
        __syncthreads();
    }
    if (threadIdx.x == 0) {
        atomicAdd(&acc[1], rg[0]);
        atomicAdd(&acc[2], rj[0]);
    }
}

// -----------------------------------------------------------------------------
// Inverse consistency: composed = fwd + trilinear_warp(bwd, fwd); sum(composed^2).
// bwd (24 MB) is L2-resident, so the 24 gathers/voxel mostly hit L2.
// -----------------------------------------------------------------------------
__global__ __launch_bounds__(256) void k_invcons(const float* __restrict__ fwd,
                                                 const float* __restrict__ bwd,
                                                 float* __restrict__ acc) {
    const int v = blockIdx.x * 256 + threadIdx.x;
    const int x = v & 127;
    const int y = (v >> 7) & 127;
    const int z = v >> 14;

    const float f0 = fwd[v];
    const float f1 = fwd[N_VOX + v];
    const float f2 = fwd[2 * N_VOX + v];

    const float cz = fminf(fmaxf((float)z + f0, 0.0f), 127.0f);
    const float cy = fminf(fmaxf((float)y + f1, 0.0f), 127.0f);
    const float cx = fminf(fmaxf((float)x + f2, 0.0f), 127.0f);
    const float z0f = floorf(cz), y0f = floorf(cy), x0f = floorf(cx);
    const float fz = cz - z0f, fy = cy - y0f, fx = cx - x0f;
    const int z0 = (int)z0f, y0 = (int)y0f, x0 = (int)x0f;
    const int z1 = min(z0 + 1, 127), y1 = min(y0 + 1, 127), x1 = min(x0 + 1, 127);

    const int iz0 = z0 << 14, iz1 = z1 << 14;
    const int iy0 = y0 << 7,  iy1 = y1 << 7;

    float fcomp[3] = { f0, f1, f2 };
    float ssum = 0.0f;
#pragma unroll
    for (int ch = 0; ch < 3; ++ch) {
        const float* b = bwd + ch * N_VOX;
        const float c000 = b[iz0 + iy0 + x0];
        const float c001 = b[iz0 + iy0 + x1];
        const float c010 = b[iz0 + iy1 + x0];
        const float c011 = b[iz0 + iy1 + x1];
        const float c100 = b[iz1 + iy0 + x0];
        const float c101 = b[iz1 + iy0 + x1];
        const float c110 = b[iz1 + iy1 + x0];
        const float c111 = b[iz1 + iy1 + x1];
        const float c00 = c000 * (1.0f - fx) + c001 * fx;
        const float c01 = c010 * (1.0f - fx) + c011 * fx;
        const float c10 = c100 * (1.0f - fx) + c101 * fx;
        const float c11 = c110 * (1.0f - fx) + c111 * fx;
        const float c0  = c00 * (1.0f - fy) + c01 * fy;
        const float c1  = c10 * (1.0f - fy) + c11 * fy;
        const float w   = c0 * (1.0f - fz) + c1 * fz;
        const float comp = fcomp[ch] + w;
        ssum += comp * comp;
    }

    __shared__ float red[256];
    red[threadIdx.x] = ssum;
    __syncthreads();
    for (int off = 128; off > 0; off >>= 1) {
        if (threadIdx.x < off) red[threadIdx.x] += red[threadIdx.x + off];
        __syncthreads();
    }
    if (threadIdx.x == 0) atomicAdd(&acc[3], red[0]);
}

// -----------------------------------------------------------------------------
// Finalize with the reference's nan_to_num semantics.
// -----------------------------------------------------------------------------
__global__ __launch_bounds__(64) void k_finalize(const float* __restrict__ acc,
                                                 float* __restrict__ out) {
    if (threadIdx.x == 0 && blockIdx.x == 0) {
        // image term: nan_to_num(1 - mean, nan=1, posinf=1, neginf=1)
        float t_img = 1.0f - acc[0] / (float)N_VOX;
        if (!isfinite(t_img)) t_img = 1.0f;

        // smoothness: (mean(dz^2)+mean(dy^2)+mean(dx^2))/3; all three means share N
        const float Ng = 3.0f * 127.0f * 128.0f * 128.0f;
        const float t_grad = acc[1] / (3.0f * Ng);

        const float t_jac = acc[2] / (127.0f * 127.0f * 127.0f);

        // inverse consistency: nan_to_num(mean, nan=0, posinf=1000, neginf=0)
        float t_inv = acc[3] / (3.0f * (float)N_VOX);
        if (isnan(t_inv)) t_inv = 0.0f;
        else if (isinf(t_inv)) t_inv = t_inv > 0.0f ? 1000.0f : 0.0f;

        out[0] = 1.0f * t_img + 0.02f * t_grad + 0.01f * t_jac + 0.1f * t_inv;
    }
}

// -----------------------------------------------------------------------------
extern "C" void kernel_launch(void* const* d_in, const int* in_sizes, int n_in,
                              void* d_out, int out_size, void* d_ws, size_t ws_size,
                              hipStream_t stream) {
    const float* src = (const float*)d_in[0];
    const float* tgt = (const float*)d_in[1];
    const float* fwd = (const float*)d_in[2];
    const float* bwd = (const float*)d_in[3];
    float* out = (float*)d_out;

    float* wsf = (float*)d_ws;
    float* acc = wsf;                         // 64 floats
    float* xf  = wsf + 64;                    // 5 * N_VOX
    float* yf  = xf + 5 * (size_t)N_VOX;      // 5 * N_VOX

    k_init<<<1, 64, 0, stream>>>(acc);

    // LNCC: x-filter on WMMA pipe (8192 wave-tiles / 8 waves per block)
    k_lncc_xfilter_wmma<<<1024, 256, 0, stream>>>(src, tgt, xf);
    k_lncc_yfilter<<<(5 * N_VOX) / 256, 256, 0, stream>>>(xf, yf);
    k_lncc_final<<<N_VOX / 256, 256, 0, stream>>>(yf, acc);

    k_gradjac<<<N_VOX / 256, 256, 0, stream>>>(fwd, acc);
    k_invcons<<<N_VOX / 256, 256, 0, stream>>>(fwd, bwd, acc);

    k_finalize<<<1, 64, 0, stream>>>(acc, out);
}